// DeBruijnGNN_5961414607058
// MI455X (gfx1250) — compile-verified
//
#include <hip/hip_runtime.h>
#include <math.h>

typedef __attribute__((ext_vector_type(2))) float v2f;
typedef __attribute__((ext_vector_type(8))) float v8f;

#define WAVE 32

__device__ __forceinline__ void atomicAddF(float* p, float v) {
    // Lowers to global_atomic_add_f32 (no CAS loop) on gfx1250.
    __hip_atomic_fetch_add(p, v, __ATOMIC_RELAXED, __HIP_MEMORY_SCOPE_AGENT);
}

// ---------------- degree / normalization ----------------

__global__ __launch_bounds__(256) void k_init_deg(int* __restrict__ deg, int n) {
    int i = blockIdx.x * blockDim.x + threadIdx.x;
    if (i < n) deg[i] = 1;   // self-loop
}

__global__ __launch_bounds__(256) void k_count_deg(int* __restrict__ deg,
                                                   const int* __restrict__ dst, int e) {
    int i = blockIdx.x * blockDim.x + threadIdx.x;
    if (i < e) atomicAdd(&deg[dst[i]], 1);
}

__global__ __launch_bounds__(256) void k_deg_to_dinv(float* __restrict__ dinv, int n) {
    int i = blockIdx.x * blockDim.x + threadIdx.x;
    if (i < n) {
        int d = ((const int*)dinv)[i];
        dinv[i] = rsqrtf((float)d);
    }
}

// ---------------- WMMA GEMM: OUT[M x (NT*16)] = X[M x 128] * W[128 x (NT*16)] ----------------

template <int NT>
__global__ __launch_bounds__(256) void k_gemm_wmma(const float* __restrict__ X,
                                                   const float* __restrict__ W,
                                                   float* __restrict__ OUT, int M) {
    constexpr int K = 128;
    constexpr int Nout = NT * 16;
    const int wave = (blockIdx.x * blockDim.x + threadIdx.x) >> 5;
    const int lane = threadIdx.x & 31;
    const int row0 = wave * 16;
    if (row0 >= M) return;                 // uniform per wave: EXEC all-1s inside

    const int mrow = row0 + (lane & 15);
    const int koff = (lane >> 4) << 1;     // 0 for lanes 0-15, 2 for lanes 16-31

    v8f acc[NT];
    const v8f zero = {0.f, 0.f, 0.f, 0.f, 0.f, 0.f, 0.f, 0.f};
#pragma unroll
    for (int t = 0; t < NT; ++t) acc[t] = zero;

    const float* xrow = X + (size_t)mrow * K;
#pragma unroll 4
    for (int k = 0; k < K; k += 4) {
        // A 16x4 fp32 fragment: VGPR0 = K=(koff), VGPR1 = K=(koff+1)
        v2f a = *(const v2f*)(xrow + k + koff);
#pragma unroll
        for (int t = 0; t < NT; ++t) {
            const int n = t * 16 + (lane & 15);
            v2f b;
            b.x = W[(size_t)(k + koff) * Nout + n];
            b.y = W[(size_t)(k + koff + 1) * Nout + n];
            acc[t] = __builtin_amdgcn_wmma_f32_16x16x4_f32(
                false, a, false, b, (short)0, acc[t], false, false);
        }
    }

    // C/D layout: VGPR i -> row (i + 8*(lane>=16)), col = lane&15
    const int nlo = lane & 15;
    const int mbase = row0 + ((lane >> 4) << 3);
#pragma unroll
    for (int t = 0; t < NT; ++t) {
#pragma unroll
        for (int i = 0; i < 8; ++i) {
            OUT[(size_t)(mbase + i) * Nout + t * 16 + nlo] = acc[t][i];
        }
    }
}

// ---------------- aggregation ----------------

// agg[n][d] = h[n][d] * dinv[n]^2   (self-loop term, also initializes buffer)
template <int LOG2D>
__global__ __launch_bounds__(256) void k_selfloop_init(const float* __restrict__ h,
                                                       const float* __restrict__ dinv,
                                                       float* __restrict__ agg, int total) {
    int i = blockIdx.x * blockDim.x + threadIdx.x;
    if (i < total) {
        int n = i >> LOG2D;
        float dv = dinv[n];
        agg[i] = h[i] * dv * dv;
    }
}

// one wave per edge: agg[dst] += h[src] * dinv[src]*dinv[dst]
template <int D>
__global__ __launch_bounds__(256) void k_edge_scatter(const float* __restrict__ h,
                                                      const int* __restrict__ src,
                                                      const int* __restrict__ dst,
                                                      const float* __restrict__ dinv,
                                                      float* __restrict__ agg, int e) {
    const int wave = (blockIdx.x * blockDim.x + threadIdx.x) >> 5;
    const int lane = threadIdx.x & 31;
    if (wave >= e) return;
    const int s = src[wave];
    const int d = dst[wave];
    const float norm = dinv[s] * dinv[d];
    constexpr int DV = D / 32;
    const float* hp = h + (size_t)s * D + lane * DV;
    float* ap = agg + (size_t)d * D + lane * DV;
#pragma unroll
    for (int j = 0; j < DV; ++j) {
        atomicAddF(ap + j, hp[j] * norm);
    }
}

// x[n][d] = relu(x[n][d] + b[d])   (D=128, in place)
__global__ __launch_bounds__(256) void k_bias_relu128(float* __restrict__ x,
                                                      const float* __restrict__ b, int total) {
    int i = blockIdx.x * blockDim.x + threadIdx.x;
    if (i < total) {
        float v = x[i] + b[i & 127];
        x[i] = fmaxf(v, 0.0f);
    }
}

// one wave per row: o[n][:] = log_softmax(o[n][:] + b2), D_OUT = 64 -> 2 cols/lane
__global__ __launch_bounds__(256) void k_bias_logsoftmax64(float* __restrict__ o,
                                                           const float* __restrict__ b, int n) {
    const int wave = (blockIdx.x * blockDim.x + threadIdx.x) >> 5;
    const int lane = threadIdx.x & 31;
    if (wave >= n) return;
    float* row = o + (size_t)wave * 64;
    float v0 = row[lane] + b[lane];
    float v1 = row[lane + 32] + b[lane + 32];
    float m = fmaxf(v0, v1);
#pragma unroll
    for (int off = 16; off > 0; off >>= 1) m = fmaxf(m, __shfl_xor(m, off, WAVE));
    float s = expf(v0 - m) + expf(v1 - m);
#pragma unroll
    for (int off = 16; off > 0; off >>= 1) s += __shfl_xor(s, off, WAVE);
    const float ls = logf(s);
    row[lane] = v0 - m - ls;
    row[lane + 32] = v1 - m - ls;
}

// ---------------- launcher ----------------

extern "C" void kernel_launch(void* const* d_in, const int* in_sizes, int n_in,
                              void* d_out, int out_size, void* d_ws, size_t ws_size,
                              hipStream_t stream) {
    const float* x  = (const float*)d_in[0];
    const int*   ei = (const int*)d_in[1];
    const float* W1 = (const float*)d_in[2];
    const float* b1 = (const float*)d_in[3];
    const float* W2 = (const float*)d_in[4];
    const float* b2 = (const float*)d_in[5];
    float* out = (float*)d_out;

    const int N = in_sizes[0] / 128;   // 100000
    const int E = in_sizes[1] / 2;     // 1600000
    const int* src = ei;
    const int* dst = ei + E;

    // workspace layout (floats): [dinv (N, int-aliased during count)] [h N*128] [agg1 N*128]
    float* ws = (float*)d_ws;
    const size_t nAlign = ((size_t)N + 255) & ~(size_t)255;
    float* dinv = ws;
    int*   deg  = (int*)ws;
    float* h    = ws + nAlign;
    float* agg1 = h + (size_t)N * 128;
    float* opre = h;                   // reuse h buffer for layer-2 pre-activation

    const int T = 256;
    const int bN      = (N + T - 1) / T;
    const int bE      = (E + T - 1) / T;
    const int bEwave  = (E + (T / WAVE) - 1) / (T / WAVE);       // wave per edge
    const int strips  = (N + 15) / 16;
    const int bGemm   = (strips + (T / WAVE) - 1) / (T / WAVE);  // wave per 16-row strip
    const int bND128  = ((size_t)N * 128 + T - 1) / T;
    const int bND64   = ((size_t)N * 64 + T - 1) / T;
    const int bRow    = (N + (T / WAVE) - 1) / (T / WAVE);       // wave per row

    // degree -> dinv
    k_init_deg<<<bN, T, 0, stream>>>(deg, N);
    k_count_deg<<<bE, T, 0, stream>>>(deg, dst, E);
    k_deg_to_dinv<<<bN, T, 0, stream>>>(dinv, N);

    // layer 1: h = x @ W1 ; agg1 = Dsym(A+I)Dsym h ; relu(+b1)
    k_gemm_wmma<8><<<bGemm, T, 0, stream>>>(x, W1, h, N);
    k_selfloop_init<7><<<bND128, T, 0, stream>>>(h, dinv, agg1, N * 128);
    k_edge_scatter<128><<<bEwave, T, 0, stream>>>(h, src, dst, dinv, agg1, E);
    k_bias_relu128<<<bND128, T, 0, stream>>>(agg1, b1, N * 128);

    // layer 2: opre = agg1 @ W2 ; out = Dsym(A+I)Dsym opre ; log_softmax(+b2)
    k_gemm_wmma<4><<<bGemm, T, 0, stream>>>(agg1, W2, opre, N);
    k_selfloop_init<6><<<bND64, T, 0, stream>>>(opre, dinv, out, N * 64);
    k_edge_scatter<64><<<bEwave, T, 0, stream>>>(opre, src, dst, dinv, out, E);
    k_bias_logsoftmax64<<<bRow, T, 0, stream>>>(out, b2, N);
}